// Net_26998164423295
// MI455X (gfx1250) — compile-verified
//
#include <hip/hip_runtime.h>
#include <math.h>

#define B_N 16384
#define F_N 48
#define T_N 512
#define L_N 16
#define NRT (B_N / 16)   // 1024 row tiles
#define NKT_F (F_N / 4)  // 12 k-tiles of idft
#define NCT_T (T_N / 16) // 32 col tiles of idft
#define NCT_F (F_N / 16) // 3 col tiles of dft

typedef __attribute__((ext_vector_type(2))) float v2f;
typedef __attribute__((ext_vector_type(8))) float v8f;

static __device__ __forceinline__ v8f wmma_f32(v2f a, v2f b, v8f c) {
  // V_WMMA_F32_16X16X4_F32 : D = A(16x4) * B(4x16) + C(16x16), full fp32
  return __builtin_amdgcn_wmma_f32_16x16x4_f32(false, a, false, b, (short)0, c,
                                               false, false);
}

// Packed WMMA-operand index: element (k, col) of a KxN matrix, tiled 4x16.
// lane = ((k>>1)&1)*16 + (col&15), comp = k&1, tile = (k>>2, col>>4).
static __device__ __forceinline__ int pack_idx(int k, int col, int nct) {
  return (((k >> 2) * nct + (col >> 4)) * 32 + ((k >> 1) & 1) * 16 + (col & 15)) * 2 +
         (k & 1);
}

// ---------------------------------------------------------------------------
// Stage 0: global max of u
// ---------------------------------------------------------------------------
__global__ void __launch_bounds__(256) k_maxpart(const float* __restrict__ u,
                                                 float* __restrict__ part, int n) {
  __shared__ float sm[256];
  float m = -3.4e38f;
  for (int i = blockIdx.x * 256 + threadIdx.x; i < n; i += 256 * gridDim.x)
    m = fmaxf(m, u[i]);
  sm[threadIdx.x] = m;
  __syncthreads();
  for (int s = 128; s > 0; s >>= 1) {
    if (threadIdx.x < s) sm[threadIdx.x] = fmaxf(sm[threadIdx.x], sm[threadIdx.x + s]);
    __syncthreads();
  }
  if (threadIdx.x == 0) part[blockIdx.x] = sm[0];
}

__global__ void __launch_bounds__(256) k_finalize(const float* __restrict__ part,
                                                  float* __restrict__ scal,
                                                  float* __restrict__ sum) {
  __shared__ float sm[256];
  float m = -3.4e38f;
  for (int i = threadIdx.x; i < 1024; i += 256) m = fmaxf(m, part[i]);
  sm[threadIdx.x] = m;
  __syncthreads();
  for (int s = 128; s > 0; s >>= 1) {
    if (threadIdx.x < s) sm[threadIdx.x] = fmaxf(sm[threadIdx.x], sm[threadIdx.x + s]);
    __syncthreads();
  }
  if (threadIdx.x == 0) {
    float mx = fabsf(sm[0]);          // abs(u.max())
    scal[0] = 5.0f / mx;              // s
    scal[1] = mx / 5.0f;              // inv
  }
  for (int t = threadIdx.x; t < L_N * T_N; t += 256) sum[t] = 0.0f;
}

// ---------------------------------------------------------------------------
// Stage 1: DFT matrices emitted directly in WMMA operand-packed order
// ---------------------------------------------------------------------------
__global__ void __launch_bounds__(256) k_dft(float* __restrict__ irep, float* __restrict__ iimp,
                                             float* __restrict__ drep, float* __restrict__ dimp) {
  int idx = blockIdx.x * 256 + threadIdx.x;
  if (idx >= F_N * T_N) return;
  int f = idx / T_N, t = idx - f * T_N;
  int ph = (f * t) & (T_N - 1);
  float ang = (float)ph * (6.283185307179586f / (float)T_N);
  const float sc = 0.04419417382415922f;  // 1/sqrt(512)
  float sv, cv;
  sincosf(ang, &sv, &cv);
  int ip = pack_idx(f, t, NCT_T);         // idft: K=48(f), N=512(t)
  irep[ip] = cv * sc;
  iimp[ip] = sv * sc;
  int dp = pack_idx(t, f, NCT_F);         // dft: K=512(t), N=48(f)
  drep[dp] = cv * sc;
  dimp[dp] = -sv * sc;
}

__global__ void __launch_bounds__(256) k_zero4(float4* __restrict__ p, int n4) {
  for (int i = blockIdx.x * 256 + threadIdx.x; i < n4; i += 256 * gridDim.x)
    p[i] = make_float4(0.f, 0.f, 0.f, 0.f);
}

// z initialized directly in packed A-operand order (per row-tile)
__global__ void __launch_bounds__(256) k_zinit(const float* __restrict__ u,
                                               const float* __restrict__ scal,
                                               float* __restrict__ zre,
                                               float* __restrict__ zim, int n) {
  const float s = scal[0];
  for (int i = blockIdx.x * 256 + threadIdx.x; i < n; i += 256 * gridDim.x) {
    int m = i / F_N, c = i - m * F_N;
    int p = (((m >> 4) * NKT_F + (c >> 2)) * 32 + ((c >> 1) & 1) * 16 + (m & 15)) * 2 +
            (c & 1);
    zre[p] = s * u[2 * i];
    zim[p] = s * u[2 * i + 1];
  }
}

// ---------------------------------------------------------------------------
// Kernel A: R = H + z @ idft  (in-place into H), accumulate sum_col(|R|^2)
// grid (32, 16) x 256; each wave: one col-tile, 8 row-tiles.
// Zero-init accumulators + late H add (H loads overlap all 48 WMMAs);
// 2-deep ping-pong prefetch of packed A operands; immediate-offset addressing.
// ---------------------------------------------------------------------------
typedef struct { v2f re[NKT_F], im[NKT_F]; } Abuf;

__global__ void __launch_bounds__(256) k_layerA(
    const float* __restrict__ zre, const float* __restrict__ zim,   // packed A
    float* __restrict__ hre, float* __restrict__ him,
    const float* __restrict__ brep, const float* __restrict__ bimp, // packed B
    float* __restrict__ sum) {
  const int lane = threadIdx.x & 31;
  const int wv = threadIdx.x >> 5;
  const int hi = lane >> 4;
  const int ct = blockIdx.x;
  const int col = ct * 16 + (lane & 15);
  const int wgl = blockIdx.y * 8 + wv;          // 0..127

  // hoist the 12 B-operand tile pairs (coalesced b64, immediate offsets)
  v2f Bre[NKT_F], Bim[NKT_F];
  {
    const float* bbr = brep + ct * 64 + lane * 2;
    const float* bbi = bimp + ct * 64 + lane * 2;
#pragma unroll
    for (int kt = 0; kt < NKT_F; ++kt) {
      Bre[kt] = *(const v2f*)(bbr + kt * (NCT_T * 64));
      Bim[kt] = *(const v2f*)(bbi + kt * (NCT_T * 64));
    }
  }

  const float* zr = zre + wgl * (8 * NKT_F * 64) + lane * 2;
  const float* zi = zim + wgl * (8 * NKT_F * 64) + lane * 2;
  float* hbr = hre + wgl * 128 * T_N + col;
  float* hbi = him + wgl * 128 * T_N + col;
  const int hrow = hi * 8 * T_N;                // row-half offset

#define LOADA(dst, j_)                                                        \
  do {                                                                        \
    _Pragma("unroll") for (int kt = 0; kt < NKT_F; ++kt) {                    \
      (dst).re[kt] = *(const v2f*)(zr + (j_) * (NKT_F * 64) + kt * 64);       \
      (dst).im[kt] = *(const v2f*)(zi + (j_) * (NKT_F * 64) + kt * 64);       \
    }                                                                         \
  } while (0)

#define TILE(A_, j_)                                                          \
  do {                                                                        \
    float hldr[8], hldi[8];                                                   \
    _Pragma("unroll") for (int i = 0; i < 8; ++i) {                           \
      hldr[i] = hbr[(j_) * (16 * T_N) + hrow + i * T_N];                      \
      hldi[i] = hbi[(j_) * (16 * T_N) + hrow + i * T_N];                      \
    }                                                                         \
    v8f ar_ = {}, ai_ = {};                                                   \
    _Pragma("unroll") for (int kt = 0; kt < NKT_F; ++kt) {                    \
      v2f bn = -Bim[kt];                                                      \
      ar_ = wmma_f32((A_).re[kt], Bre[kt], ar_);                              \
      ar_ = wmma_f32((A_).im[kt], bn, ar_);                                   \
      ai_ = wmma_f32((A_).re[kt], Bim[kt], ai_);                              \
      ai_ = wmma_f32((A_).im[kt], Bre[kt], ai_);                              \
    }                                                                         \
    _Pragma("unroll") for (int i = 0; i < 8; ++i) {                           \
      float rr = ar_[i] + hldr[i], ri = ai_[i] + hldi[i];                     \
      hbr[(j_) * (16 * T_N) + hrow + i * T_N] = rr;                           \
      hbi[(j_) * (16 * T_N) + hrow + i * T_N] = ri;                           \
      colsum += rr * rr + ri * ri;                                            \
    }                                                                         \
  } while (0)

  float colsum = 0.0f;
  Abuf A0, A1;
  LOADA(A0, 0);
  for (int jj = 0; jj < 8; jj += 2) {
    LOADA(A1, jj + 1);                          // prefetch odd tile
    TILE(A0, jj);
    if (jj + 2 < 8) LOADA(A0, jj + 2);          // prefetch next even tile
    TILE(A1, jj + 1);
  }
#undef LOADA
#undef TILE

  colsum += __shfl_xor(colsum, 16, 32);         // lanes l, l+16 share a column
  if (lane < 16) atomicAdd(&sum[col], colsum);
}

// ---------------------------------------------------------------------------
// Kernel C: g = sigmoid(S1*(sqrt(sum/B)-S2)); H = R*g (in place);
//           h = H @ dft;  z = s*u - h  (packed) or out = h*inv on last layer.
// grid (1024) x 256; block = one 16-row strip; 8 waves split K=512 (64 each),
// partial 16x16 f32 tiles reduced through LDS ds_add_f32.
// ---------------------------------------------------------------------------
__global__ void __launch_bounds__(256) k_layerC(
    float* __restrict__ hre, float* __restrict__ him,
    const float* __restrict__ drep, const float* __restrict__ dimp, // packed B
    float* __restrict__ zre, float* __restrict__ zim,               // packed A out
    const float* __restrict__ u, const float* __restrict__ scal,
    const float* __restrict__ sum, const float* __restrict__ S1,
    const float* __restrict__ S2, float* __restrict__ out, int last) {
  __shared__ float gs[T_N];
  __shared__ float accL[6][256];                // 3 n-tiles x {re,im}, 16x16 each
  const float s1 = S1[0], s2 = S2[0];
  for (int t = threadIdx.x; t < T_N; t += 256) {
    float rm = sqrtf(sum[t] * (1.0f / (float)B_N));
    gs[t] = 1.0f / (1.0f + expf(-(s1 * (rm - s2))));
  }
  for (int i = threadIdx.x; i < 6 * 256; i += 256) (&accL[0][0])[i] = 0.0f;
  __syncthreads();

  const int lane = threadIdx.x & 31;
  const int wv = threadIdx.x >> 5;              // this wave's K-chunk: [wv*64, wv*64+64)
  const int rowbase = blockIdx.x * 16;
  const int kbase = wv * 64;

  // Phase 1: gate this wave's own K-chunk of the strip, in place (float4).
  for (int j = 0; j < 8; ++j) {
    int idx = (j * 32 + lane) * 4;              // 0..1023 within 16x64 chunk
    int row = idx >> 6, cc = idx & 63;
    int col = kbase + cc;
    float4 g4 = *(const float4*)&gs[col];
    int off = (rowbase + row) * T_N + col;
    float4 r = *(float4*)&hre[off];
    r.x *= g4.x; r.y *= g4.y; r.z *= g4.z; r.w *= g4.w;
    *(float4*)&hre[off] = r;
    float4 q = *(float4*)&him[off];
    q.x *= g4.x; q.y *= g4.y; q.z *= g4.z; q.w *= g4.w;
    *(float4*)&him[off] = q;
  }

  // Phase 2: partial h over this wave's K-chunk (same chunk it just gated:
  // same-wave store->load ordering is HW guaranteed). 2-step SW pipeline.
  v8f accRe[3] = {}, accIm[3] = {};
  const int hi = lane >> 4;
  const int cloc = lane & 15;
  const float* harp = hre + (rowbase + cloc) * T_N + kbase + hi * 2;
  const float* haip = him + (rowbase + cloc) * T_N + kbase + hi * 2;
  const float* dbr = drep + (kbase >> 2) * (NCT_F * 64) + lane * 2;
  const float* dbi = dimp + (kbase >> 2) * (NCT_F * 64) + lane * 2;

  v2f pr[2], pi[2];
  pr[0] = *(const v2f*)(harp + 0);
  pi[0] = *(const v2f*)(haip + 0);
  pr[1] = *(const v2f*)(harp + 4);
  pi[1] = *(const v2f*)(haip + 4);
#pragma unroll
  for (int t = 0; t < 16; ++t) {
    v2f a_re = pr[t & 1], a_im = pi[t & 1];
    if (t + 2 < 16) {
      pr[t & 1] = *(const v2f*)(harp + (t + 2) * 4);
      pi[t & 1] = *(const v2f*)(haip + (t + 2) * 4);
    }
#pragma unroll
    for (int nt = 0; nt < NCT_F; ++nt) {
      v2f b_re = *(const v2f*)(dbr + (t * NCT_F + nt) * 64);
      v2f b_im = *(const v2f*)(dbi + (t * NCT_F + nt) * 64);
      v2f bn = -b_im;
      accRe[nt] = wmma_f32(a_re, b_re, accRe[nt]);
      accRe[nt] = wmma_f32(a_im, bn, accRe[nt]);
      accIm[nt] = wmma_f32(a_re, b_im, accIm[nt]);
      accIm[nt] = wmma_f32(a_im, b_re, accIm[nt]);
    }
  }

  // Reduce the 8 waves' partial tiles in LDS (ds_add_f32).
#pragma unroll
  for (int nt = 0; nt < NCT_F; ++nt) {
#pragma unroll
    for (int i = 0; i < 8; ++i) {
      int sl = (i + hi * 8) * 16 + cloc;
      atomicAdd(&accL[nt * 2 + 0][sl], accRe[nt][i]);
      atomicAdd(&accL[nt * 2 + 1][sl], accIm[nt][i]);
    }
  }
  __syncthreads();

  // Epilogue: thread-linear over the 16x48 strip.
  const float s = scal[0], inv = scal[1];
  const int rt = blockIdx.x;
#pragma unroll
  for (int ii = 0; ii < 3; ++ii) {
    int e = ii * 256 + threadIdx.x;             // 0..767
    int row = e / F_N, c = e - row * F_N;
    int m = rowbase + row;
    float hr = accL[(c >> 4) * 2 + 0][row * 16 + (c & 15)];
    float hm = accL[(c >> 4) * 2 + 1][row * 16 + (c & 15)];
    if (last) {
      out[(m * F_N + c) * 2]     = hr * inv;
      out[(m * F_N + c) * 2 + 1] = hm * inv;
    } else {
      int p = ((rt * NKT_F + (c >> 2)) * 32 + ((c >> 1) & 1) * 16 + row) * 2 + (c & 1);
      zre[p] = s * u[(m * F_N + c) * 2]     - hr;
      zim[p] = s * u[(m * F_N + c) * 2 + 1] - hm;
    }
  }
}

// ---------------------------------------------------------------------------
extern "C" void kernel_launch(void* const* d_in, const int* in_sizes, int n_in,
                              void* d_out, int out_size, void* d_ws, size_t ws_size,
                              hipStream_t stream) {
  const float* u  = (const float*)d_in[0];
  const float* S1 = (const float*)d_in[1];
  const float* S2 = (const float*)d_in[2];
  float* out = (float*)d_out;
  float* w = (float*)d_ws;

  // workspace layout (float offsets)
  float* scal = w;                  // 8
  float* part = w + 8;              // 1024
  float* sum  = w + 2048;           // 16*512
  float* irep = w + 16384;          // 48*512 packed
  float* iimp = irep + F_N * T_N;
  float* drep = iimp + F_N * T_N;
  float* dimp = drep + F_N * T_N;
  float* zre  = w + 131072;         // B*F packed
  float* zim  = zre + B_N * F_N;
  float* hre  = zim + B_N * F_N;    // B*T row-major (contiguous with him)
  float* him  = hre + B_N * T_N;    // total ~70.5 MiB

  const int nu = B_N * F_N * 2;
  k_maxpart<<<1024, 256, 0, stream>>>(u, part, nu);
  k_finalize<<<1, 256, 0, stream>>>(part, scal, sum);
  k_dft<<<(F_N * T_N + 255) / 256, 256, 0, stream>>>(irep, iimp, drep, dimp);
  k_zero4<<<4096, 256, 0, stream>>>((float4*)hre, (2 * B_N * T_N) / 4);
  k_zinit<<<1024, 256, 0, stream>>>(u, scal, zre, zim, B_N * F_N);

  for (int l = 0; l < L_N; ++l) {
    k_layerA<<<dim3(NCT_T, 16), 256, 0, stream>>>(zre, zim, hre, him, irep, iimp,
                                                  sum + l * T_N);
    k_layerC<<<NRT, 256, 0, stream>>>(hre, him, drep, dimp, zre, zim, u, scal,
                                      sum + l * T_N, S1, S2, out,
                                      (l == L_N - 1) ? 1 : 0);
  }
}